// BTTMoELayer_18279380812216
// MI455X (gfx1250) — compile-verified
//
#include <hip/hip_runtime.h>
#include <math.h>

// ---------------------------------------------------------------------------
// BTT-MoE for MI455X (gfx1250, wave32).
// N=8192 tokens, d=1024 (n1=32 x n2=32), m1=64, m2=64, E=8, top-2 gating.
// Useful compute ~6.4 GFLOP fp32 (top-2 sparsity exploited via expert gather);
// all matmuls on v_wmma_f32_16x16x4_f32 (the only f32 WMMA shape).
// Weights pre-swizzled into per-lane WMMA B-fragment order so each fragment
// is a single coalesced global_load_b64.
// ---------------------------------------------------------------------------

typedef __attribute__((ext_vector_type(2))) float v2f;
typedef __attribute__((ext_vector_type(8))) float v8f;

#define D_IN    1024
#define N1      32
#define N2      32
#define M1      64
#define M2      64
#define NEXP    8
#define DOUT    4096

// LDS pitches: 16-lane strided accesses hit 16 distinct banks.
#define XPITCH  1028          // 1028 % 64 == 4  -> banks {0,4,...,60}; 16B aligned rows
#define ZBP     33            // token-row stride in Z (33*b mod 64 distinct, b<16)
#define ZKP     529           // k stride: 16*33+1 (529*k mod 64 distinct over 16)

__device__ __forceinline__ v8f wmma4(v2f a, v2f b, v8f c) {
  // D = A(16x4,f32) * B(4x16,f32) + C(16x16,f32)
  return __builtin_amdgcn_wmma_f32_16x16x4_f32(false, a, false, b, (short)0, c,
                                               false, false);
}

// ---------------------------------------------------------------------------
// Kernel 0: out[i] = bias[i % 4096] (float4), zero the 8 expert counters.
// ---------------------------------------------------------------------------
__global__ void init_out_kernel(float4* __restrict__ out,
                                const float4* __restrict__ bias4,
                                long total4, int* __restrict__ counts) {
  long i = (long)blockIdx.x * blockDim.x + threadIdx.x;
  if (i < NEXP) counts[i] = 0;
  if (i < total4) out[i] = bias4[i & (DOUT / 4 - 1)];
}

// ---------------------------------------------------------------------------
// Kernel 1a: W1 (n2,n1,m1,E) -> fragment-swizzled W1s[e][j][s][half][ct][l15][2]
// where i = 4s + 2*half + p, k = 16*ct + l15. One b64 per WMMA B-fragment.
// ---------------------------------------------------------------------------
__global__ void w1_swz_kernel(const float* __restrict__ W1,
                              float* __restrict__ W1s) {
  int o = blockIdx.x * blockDim.x + threadIdx.x;
  if (o >= N2 * N1 * M1 * NEXP) return;
  int p    = o & 1;
  int l15  = (o >> 1) & 15;
  int ct   = (o >> 5) & 3;
  int half = (o >> 7) & 1;
  int s    = (o >> 8) & 7;
  int j    = (o >> 11) & 31;
  int e    = (o >> 16) & 7;
  int i = 4 * s + 2 * half + p;
  int k = 16 * ct + l15;
  W1s[o] = W1[(((j * N1 + i) * M1 + k) * NEXP) + e];
}

// ---------------------------------------------------------------------------
// Kernel 1b: W2 (E,m1,n2,m2) -> fragment-swizzled W2s[e][k][s][half][ct][l15][2]
// where j = 4s + 2*half + p, l = 16*ct + l15.
// ---------------------------------------------------------------------------
__global__ void w2_swz_kernel(const float* __restrict__ W2,
                              float* __restrict__ W2s) {
  int o = blockIdx.x * blockDim.x + threadIdx.x;
  if (o >= NEXP * M1 * N2 * M2) return;
  int p    = o & 1;
  int l15  = (o >> 1) & 15;
  int ct   = (o >> 5) & 3;
  int half = (o >> 7) & 1;
  int s    = (o >> 8) & 7;
  int k    = (o >> 11) & 63;
  int e    = (o >> 17) & 7;
  int j = 4 * s + 2 * half + p;
  int l = 16 * ct + l15;
  W2s[o] = W2[((long)(e * M1 + k) * N2 + j) * M2 + l];
}

// ---------------------------------------------------------------------------
// Kernel 2: gate. One wave32 per token: 8 logits, shfl_xor reduce, top-2 +
// softmax(+1e-6), dense sparse_w, expert-major token gather lists.
// ---------------------------------------------------------------------------
__global__ void __launch_bounds__(256)
gate_kernel(const float* __restrict__ x, const float* __restrict__ Wg,
            float* __restrict__ sw, int* __restrict__ counts,
            int* __restrict__ lists, int N) {
  const int token = blockIdx.x * 8 + (threadIdx.x >> 5);
  const int lane = threadIdx.x & 31;
  if (token >= N) return;

  float acc[NEXP];
#pragma unroll
  for (int e = 0; e < NEXP; ++e) acc[e] = 0.f;

  const float* xr = x + (long)token * D_IN;
  for (int t = 0; t < D_IN / 32; ++t) {
    float xv = xr[t * 32 + lane];
#pragma unroll
    for (int e = 0; e < NEXP; ++e)
      acc[e] += xv * Wg[e * D_IN + t * 32 + lane];
  }
#pragma unroll
  for (int e = 0; e < NEXP; ++e) {
#pragma unroll
    for (int off = 16; off > 0; off >>= 1)
      acc[e] += __shfl_xor(acc[e], off, 32);
  }

  if (lane == 0) {
    int i0 = 0; float v0 = acc[0];
#pragma unroll
    for (int e = 1; e < NEXP; ++e)
      if (acc[e] > v0) { v0 = acc[e]; i0 = e; }
    int i1 = -1; float v1 = -3.4e38f;
#pragma unroll
    for (int e = 0; e < NEXP; ++e)
      if (e != i0 && acc[e] > v1) { v1 = acc[e]; i1 = e; }

    float ex = __expf(v1 - v0);
    float inv = 1.f / (1.f + ex);
    float p0 = inv + 1e-6f;
    float p1 = ex * inv + 1e-6f;

    float* swr = sw + (long)token * NEXP;
#pragma unroll
    for (int e = 0; e < NEXP; ++e) swr[e] = 0.f;
    swr[i0] = p0;
    swr[i1] = p1;

    int p = atomicAdd(&counts[i0], 1);
    lists[i0 * N + p] = token;
    p = atomicAdd(&counts[i1], 1);
    lists[i1 * N + p] = token;
  }
}

// ---------------------------------------------------------------------------
// Kernel 3: main BTT compute. Grid (ceil(N/16), 8 experts), 8 waves / WG.
// Stage 1: per wave 4 j-blocks: Z_j(16x64) = X_j(16x32) @ W1[e][j](32x64),
//          row-scaled by gate weight, stored to LDS Z[k][b][j].
// Stage 2: per wave 8 k-values: Y_k(16x64) = Z_k(16x32) @ W2[e][k](32x64),
//          atomically accumulated into d_out (bias pre-loaded).
// ---------------------------------------------------------------------------
__global__ void __launch_bounds__(256)
btt_moe_main(const float* __restrict__ x, const float* __restrict__ W1s,
             const float* __restrict__ W2s, const float* __restrict__ sw,
             const int* __restrict__ counts, const int* __restrict__ lists,
             float* __restrict__ out, int N) {
  extern __shared__ float smem[];
  float* Xs  = smem;                       // 16 * XPITCH
  float* Zs  = Xs + 16 * XPITCH;           // 64 * ZKP  (Z[k*ZKP + b*ZBP + j])
  float* wls = Zs + 64 * ZKP;              // 16 gate weights
  int*   tls = (int*)(wls + 16);           // 16 token ids

  const int e = blockIdx.y;
  const int cnt = counts[e];               // uniform -> EXEC stays full
  const int tile0 = blockIdx.x * 16;
  if (tile0 >= cnt) return;

  const int tid  = threadIdx.x;
  const int lane = tid & 31;
  const int wv   = tid >> 5;               // wave id 0..7
  const int half = lane >> 4;              // 0 | 1
  const int l15  = lane & 15;

  if (tid < 16) {
    int idx = tile0 + tid;
    int t; float wt;
    if (idx < cnt) { t = lists[e * N + idx]; wt = sw[(long)t * NEXP + e]; }
    else           { t = lists[e * N + tile0]; wt = 0.f; }  // padded row: w=0
    tls[tid] = t;
    wls[tid] = wt;
  }
  __syncthreads();

  // Load 16 x 1024 token rows into LDS as float4 (b128, coalesced).
  for (int it = 0; it < (16 * D_IN / 4) / 256; ++it) {
    int idx = it * 256 + tid;
    int b = idx >> 8;                      // 256 float4 per row
    int d4 = idx & 255;
    ((float4*)(Xs + b * XPITCH))[d4] =
        ((const float4*)(x + (long)tls[b] * D_IN))[d4];
  }
  __syncthreads();

  // per-lane row scales: C/D VGPR g holds rows {g, g+8} by lane half
  float wr[8];
#pragma unroll
  for (int g = 0; g < 8; ++g) wr[g] = wls[g + 8 * half];

  // ---------------- stage 1 ----------------
  for (int jj = 0; jj < 4; ++jj) {
    const int j = wv * 4 + jj;
    const v2f* W1b = (const v2f*)(W1s + (long)(e * N2 + j) * (N1 * M1));
    v8f a0 = {0,0,0,0,0,0,0,0}, a1 = {0,0,0,0,0,0,0,0};
    v8f a2 = {0,0,0,0,0,0,0,0}, a3 = {0,0,0,0,0,0,0,0};
#pragma unroll
    for (int s = 0; s < 8; ++s) {             // K = i, 4 per step
      const int i0 = 4 * s + 2 * half;
      v2f a;                                  // A[l15][i0], A[l15][i0+1]
      a.x = Xs[l15 * XPITCH + (i0 + 0) * N2 + j];
      a.y = Xs[l15 * XPITCH + (i0 + 1) * N2 + j];
      const v2f* Wv = W1b + ((s * 2 + half) * 64 + l15);  // one b64 / fragment
      v2f b0 = Wv[0], b1 = Wv[16], b2 = Wv[32], b3 = Wv[48];
      a0 = wmma4(a, b0, a0);
      a1 = wmma4(a, b1, a1);
      a2 = wmma4(a, b2, a2);
      a3 = wmma4(a, b3, a3);
    }
    // scale rows by gate weight, store Z[k][b][j]
#pragma unroll
    for (int g = 0; g < 8; ++g) {
      const int brow = g + 8 * half;
      float* zb = Zs + (long)brow * ZBP + j;
      zb[( 0 + l15) * ZKP] = a0[g] * wr[g];
      zb[(16 + l15) * ZKP] = a1[g] * wr[g];
      zb[(32 + l15) * ZKP] = a2[g] * wr[g];
      zb[(48 + l15) * ZKP] = a3[g] * wr[g];
    }
  }
  __syncthreads();

  // ---------------- stage 2 ----------------
  for (int kk = 0; kk < 8; ++kk) {
    const int k = wv * 8 + kk;
    const v2f* W2b = (const v2f*)(W2s + (long)(e * M1 + k) * (N2 * M2));
    const float* Za = Zs + (long)k * ZKP + l15 * ZBP;
    v8f a0 = {0,0,0,0,0,0,0,0}, a1 = {0,0,0,0,0,0,0,0};
    v8f a2 = {0,0,0,0,0,0,0,0}, a3 = {0,0,0,0,0,0,0,0};
#pragma unroll
    for (int s = 0; s < 8; ++s) {             // K = j, 4 per step
      const int j0 = 4 * s + 2 * half;
      v2f a;
      a.x = Za[j0];
      a.y = Za[j0 + 1];
      const v2f* Wv = W2b + ((s * 2 + half) * 64 + l15);  // one b64 / fragment
      v2f b0 = Wv[0], b1 = Wv[16], b2 = Wv[32], b3 = Wv[48];
      a0 = wmma4(a, b0, a0);
      a1 = wmma4(a, b1, a1);
      a2 = wmma4(a, b2, a2);
      a3 = wmma4(a, b3, a3);
    }
#pragma unroll
    for (int g = 0; g < 8; ++g) {
      const int brow = g + 8 * half;
      float* ob = out + (long)tls[brow] * DOUT + k * M2 + l15;
      atomicAdd(ob +  0, a0[g]);
      atomicAdd(ob + 16, a1[g]);
      atomicAdd(ob + 32, a2[g]);
      atomicAdd(ob + 48, a3[g]);
    }
  }
}

// ---------------------------------------------------------------------------
extern "C" void kernel_launch(void* const* d_in, const int* in_sizes, int n_in,
                              void* d_out, int out_size, void* d_ws,
                              size_t ws_size, hipStream_t stream) {
  const float* x  = (const float*)d_in[0];
  const float* Wg = (const float*)d_in[1];
  const float* W1 = (const float*)d_in[2];
  const float* W2 = (const float*)d_in[3];
  const float* b  = (const float*)d_in[4];
  float* out = (float*)d_out;

  const int N = in_sizes[0] / D_IN;        // 8192 tokens

  // workspace layout (all 16B-aligned chunks)
  float* sw   = (float*)d_ws;                       // N*8 gate weights
  int* counts = (int*)(sw + (long)N * NEXP);        // 8
  int* lists  = counts + NEXP + 8;                  // 8*N token ids (+pad align)
  float* W1s  = (float*)(lists + (long)NEXP * N);   // 2 MB swizzled W1
  float* W2s  = W1s + (long)NEXP * N2 * N1 * M1;    // 4 MB swizzled W2

  const long total4 = (long)N * DOUT / 4;
  init_out_kernel<<<(int)((total4 + 255) / 256), 256, 0, stream>>>(
      (float4*)out, (const float4*)b, total4, counts);

  const int w1n = N2 * N1 * M1 * NEXP;
  w1_swz_kernel<<<(w1n + 255) / 256, 256, 0, stream>>>(W1, W1s);
  const int w2n = NEXP * M1 * N2 * M2;
  w2_swz_kernel<<<(w2n + 255) / 256, 256, 0, stream>>>(W2, W2s);

  gate_kernel<<<N / 8, 256, 0, stream>>>(x, Wg, sw, counts, lists, N);

  const size_t smem = (size_t)(16 * XPITCH + 64 * ZKP + 16) * sizeof(float) +
                      16 * sizeof(int);    // ~201 KB of the 320 KB WGP LDS
  dim3 grid((N + 15) / 16, NEXP);
  btt_moe_main<<<grid, 256, smem, stream>>>(x, W1s, W2s, sw, counts, lists, out,
                                            N);
}